// GAT_4861902979239
// MI455X (gfx1250) — compile-verified
//
#include <hip/hip_runtime.h>
#include <cstdint>
#include <cstddef>

typedef float v2f __attribute__((ext_vector_type(2)));
typedef float v4f __attribute__((ext_vector_type(4)));
typedef float v8f __attribute__((ext_vector_type(8)));

#define NN 8192
#define FD 128
#define KC 64            // columns of adj processed per chunk
#define PST 66           // padded LDS stride for P tile (conflict-free, 8B-align)
#define MB 32            // output rows per block (2 M-tiles of 16)
#define PFC 8            // prefetch distance in chunks

// ---------------------------------------------------------------------------
// Kernel 1: x' = x @ W + bias, plus per-row scores s_src = x'·phi[:128],
//           s_dst = x'·phi[128:]. One block handles 16 rows.
// ---------------------------------------------------------------------------
__global__ __launch_bounds__(256) void gat_xprime_kernel(
    const float* __restrict__ x, const float* __restrict__ W,
    const float* __restrict__ bias, const float* __restrict__ phi,
    float* __restrict__ xp, float* __restrict__ s_src, float* __restrict__ s_dst)
{
    __shared__ float Ws[FD * FD];     // 64 KB  (LDS per WGP is 320 KB)
    __shared__ float xs[16 * FD];     // 8 KB
    __shared__ float red[256];

    const int tid = threadIdx.x;
    const int i0  = blockIdx.x * 16;

    for (int idx = tid; idx < FD * FD; idx += 256) Ws[idx] = W[idx];
    for (int idx = tid; idx < 16 * FD; idx += 256) xs[idx] = x[(size_t)i0 * FD + idx];
    __syncthreads();

    const int   col = tid & 127;
    const int   rh  = tid >> 7;            // 0..1: two rows per pass
    const float phs = phi[col];
    const float phd = phi[FD + col];
    const float bv  = bias[col];

    for (int pass = 0; pass < 8; ++pass) {
        const int r = pass * 2 + rh;
        float acc = bv;
        #pragma unroll 8
        for (int k = 0; k < FD; ++k)
            acc = fmaf(xs[r * FD + k], Ws[k * FD + col], acc);
        xp[(size_t)(i0 + r) * FD + col] = acc;

        red[tid] = acc * phs;
        __syncthreads();
        for (int s = 64; s > 0; s >>= 1) {
            if ((tid & 127) < s) red[tid] += red[tid + s];
            __syncthreads();
        }
        if ((tid & 127) == 0) s_src[i0 + r] = red[tid];
        __syncthreads();

        red[tid] = acc * phd;
        __syncthreads();
        for (int s = 64; s > 0; s >>= 1) {
            if ((tid & 127) < s) red[tid] += red[tid + s];
            __syncthreads();
        }
        if ((tid & 127) == 0) s_dst[i0 + r] = red[tid];
        __syncthreads();
    }
}

// ---------------------------------------------------------------------------
// Kernel 2: gmax = max_j s_dst[j]   (single workgroup)
// ---------------------------------------------------------------------------
__global__ __launch_bounds__(256) void gat_gmax_kernel(
    const float* __restrict__ s_dst, float* __restrict__ gmax)
{
    __shared__ float red[256];
    const int tid = threadIdx.x;
    float m = -3.4e38f;
    for (int j = tid; j < NN; j += 256) m = fmaxf(m, s_dst[j]);
    red[tid] = m;
    __syncthreads();
    for (int s = 128; s > 0; s >>= 1) {
        if (tid < s) red[tid] = fmaxf(red[tid], red[tid + s]);
        __syncthreads();
    }
    if (tid == 0) gmax[0] = red[0];
}

// ---------------------------------------------------------------------------
// Kernel 3: fused masked-softmax-attention @ x'  (flash-attention style).
// Block = 256 threads = 8 waves; block handles 32 output rows (two 16-row
// M-tiles per wave, so each B fragment feeds two WMMAs -> halves L2 traffic).
// Wave w owns output columns [16w, 16w+16).
// Shift = leaky(s_src_i + gmax) >= per-row max, so a single pass with a
// running row-sum suffices (no online rescaling).
// adj is a 256 MB single-use stream -> non-temporal loads + prefetch.
// ---------------------------------------------------------------------------
__global__ __launch_bounds__(256) void gat_attn_kernel(
    const float* __restrict__ adj, const float* __restrict__ xp,
    const float* __restrict__ s_src, const float* __restrict__ s_dst,
    const float* __restrict__ gmax, float* __restrict__ out)
{
    __shared__ float p_lds[MB * PST];     // P tile, padded stride (~8.4 KB)
    __shared__ float ssrc_s[MB];
    __shared__ float shift_s[MB];
    __shared__ float psum[2 * 256];
    __shared__ float rowsum[MB];

    const int tid  = threadIdx.x;
    const int wave = tid >> 5;
    const int lane = tid & 31;
    const int half = lane >> 4;           // 0/1 lane-half (WMMA layout)
    const int l15  = lane & 15;
    const int i0   = blockIdx.x * MB;
    const int n0   = wave * 16;           // this wave's output column tile

    if (tid < MB) {
        float ss = s_src[i0 + tid];
        ssrc_s[tid] = ss;
        float sh = ss + gmax[0];
        shift_s[tid] = (sh > 0.0f) ? sh : 0.2f * sh;   // leaky(row-max bound)
    }
    __syncthreads();

    // P-compute mapping: thread -> (rows r and r+16, 4 consecutive cols)
    const int r  = tid >> 4;              // 0..15
    const int c4 = (tid & 15) * 4;        // 0..60
    const float ssrc0  = ssrc_s[r];
    const float ssrc1  = ssrc_s[16 + r];
    const float shift0 = shift_s[r];
    const float shift1 = shift_s[16 + r];
    const float* adj_row0 = adj + (size_t)(i0 + r) * NN;
    const float* adj_row1 = adj + (size_t)(i0 + 16 + r) * NN;

    v8f acc0 = {0.f, 0.f, 0.f, 0.f, 0.f, 0.f, 0.f, 0.f};
    v8f acc1 = {0.f, 0.f, 0.f, 0.f, 0.f, 0.f, 0.f, 0.f};
    float lsum0 = 0.0f;
    float lsum1 = 0.0f;

    for (int jc = 0; jc < NN; jc += KC) {
        // ---- build MB x KC tile of attention numerators in LDS ----
        const v4f a40 = __builtin_nontemporal_load((const v4f*)(adj_row0 + jc + c4));
        const v4f a41 = __builtin_nontemporal_load((const v4f*)(adj_row1 + jc + c4));
        const v4f d4  = *(const v4f*)(s_dst + jc + c4);
        if (jc + PFC * KC < NN) {   // stream-ahead prefetch of the adj rows
            __builtin_prefetch(adj_row0 + jc + PFC * KC + c4, 0, 0);
            __builtin_prefetch(adj_row1 + jc + PFC * KC + c4, 0, 0);
        }
        #pragma unroll
        for (int e = 0; e < 4; ++e) {
            const int j = jc + c4 + e;
            float S0 = ssrc0 + d4[e];
            float S1 = ssrc1 + d4[e];
            S0 = (S0 > 0.0f) ? S0 : 0.2f * S0;                 // leaky_relu
            S1 = (S1 > 0.0f) ? S1 : 0.2f * S1;
            const bool k0 = (a40[e] != 0.0f) || (j == i0 + r);        // adj+I>0
            const bool k1 = (a41[e] != 0.0f) || (j == i0 + 16 + r);
            const float p0 = k0 ? __expf(S0 - shift0) : 0.0f;
            const float p1 = k1 ? __expf(S1 - shift1) : 0.0f;
            p_lds[r * PST + c4 + e]        = p0;
            p_lds[(16 + r) * PST + c4 + e] = p1;
            lsum0 += p0;
            lsum1 += p1;
        }
        __syncthreads();

        // ---- 32 x 16 x KC contraction with x' via f32 WMMA (K-step 4) ----
        #pragma unroll
        for (int kk = 0; kk < KC; kk += 4) {
            // A frags (16x4): VGPR0 = K=(2*half), VGPR1 = K=(2*half+1); M=l15
            v2f a0 = *(const v2f*)(p_lds + l15 * PST + kk + 2 * half);
            v2f a1 = *(const v2f*)(p_lds + (16 + l15) * PST + kk + 2 * half);
            // B frag (4x16): rows jc+kk+2*half(+1) of x', cols n0+l15
            const float* xb = xp + (size_t)(jc + kk + 2 * half) * FD + n0 + l15;
            v2f b;
            b.x = xb[0];
            b.y = xb[FD];
            acc0 = __builtin_amdgcn_wmma_f32_16x16x4_f32(
                false, a0, false, b, (short)0, acc0, false, false);
            acc1 = __builtin_amdgcn_wmma_f32_16x16x4_f32(
                false, a1, false, b, (short)0, acc1, false, false);
        }
        __syncthreads();
    }

    // ---- row sums: threads [16m,16m+16) belong to rows m and m+16 ----
    psum[tid]       = lsum0;
    psum[256 + tid] = lsum1;
    __syncthreads();
    if (tid < MB) {
        const int g = tid >> 4;          // 0: rows 0-15, 1: rows 16-31
        const int m = tid & 15;
        float s = 0.0f;
        #pragma unroll
        for (int q = 0; q < 16; ++q) s += psum[g * 256 + m * 16 + q];
        rowsum[tid] = s;
    }
    __syncthreads();

    // ---- writeback: C/D layout VGPR v -> M = v + 8*half, N = n0 + l15 ----
    #pragma unroll
    for (int v = 0; v < 8; ++v) {
        const int m = v + 8 * half;
        out[(size_t)(i0 + m) * FD + n0 + l15]      = acc0[v] / rowsum[m];
        out[(size_t)(i0 + 16 + m) * FD + n0 + l15] = acc1[v] / rowsum[16 + m];
    }
}

// ---------------------------------------------------------------------------
extern "C" void kernel_launch(void* const* d_in, const int* in_sizes, int n_in,
                              void* d_out, int out_size, void* d_ws, size_t ws_size,
                              hipStream_t stream)
{
    const float* adj  = (const float*)d_in[0];   // [8192, 8192]
    const float* x    = (const float*)d_in[1];   // [8192, 128]
    const float* W    = (const float*)d_in[2];   // [128, 128]
    const float* bias = (const float*)d_in[3];   // [128]
    const float* phi  = (const float*)d_in[4];   // [256, 1]
    float* out = (float*)d_out;                  // [8192, 128]

    // workspace: x' (4 MB) + s_src + s_dst + gmax  (~4.26 MB total)
    float* xp    = (float*)d_ws;
    float* s_src = xp + (size_t)NN * FD;
    float* s_dst = s_src + NN;
    float* gmax  = s_dst + NN;

    gat_xprime_kernel<<<NN / 16, 256, 0, stream>>>(x, W, bias, phi, xp, s_src, s_dst);
    gat_gmax_kernel<<<1, 256, 0, stream>>>(s_dst, gmax);
    gat_attn_kernel<<<NN / MB, 256, 0, stream>>>(adj, xp, s_src, s_dst, gmax, out);
}